// Contrastive_75496935129381
// MI455X (gfx1250) — compile-verified
//
#include <hip/hip_runtime.h>
#include <math.h>

typedef __attribute__((ext_vector_type(2))) float v2f;
typedef __attribute__((ext_vector_type(8))) float v8f;
typedef __attribute__((ext_vector_type(4))) int v4i;

#if defined(__has_builtin)
#if __has_builtin(__builtin_amdgcn_wmma_f32_16x16x4_f32)
#define HAVE_WMMA_F32 1
#endif
#if __has_builtin(__builtin_amdgcn_global_load_async_to_lds_b128)
#define HAVE_ASYNC 1
#endif
#endif
#ifndef HAVE_WMMA_F32
#define HAVE_WMMA_F32 0
#endif
#ifndef HAVE_ASYNC
#define HAVE_ASYNC 0
#endif

#define B_ 64
#define C_ 1024
#define DT_ 1024          // D*T
#define M_ 65536          // B*DT
#define N_ 1024           // 4*F4
#define VN 50
#define LDA_P 72          // proj [k][m] stride: 2*72 % 64 == 16 -> half-wave bank split
#define LDA_S 44          // sim [row][k] stride: 16B aligned, banks {0,1}/{2,3} mod 4 split

// 16B global->LDS copy: async direct-to-LDS on CDNA5, sync fallback otherwise.
__device__ __forceinline__ void cp128(const float* g, float* l) {
#if HAVE_ASYNC
  __builtin_amdgcn_global_load_async_to_lds_b128(
      (__attribute__((address_space(1))) v4i*)(g),
      (__attribute__((address_space(3))) v4i*)(l), 0, 0);
#else
  *(float4*)l = *(const float4*)g;
#endif
}

__device__ __forceinline__ void wait_async_lds() {
#if HAVE_ASYNC
  asm volatile("s_wait_asynccnt 0" ::: "memory");
#endif
}

// ---------------- small prep kernels ----------------

__global__ void k_vmean(const float* __restrict__ vid, float* __restrict__ vmean) {
  int m = blockIdx.x * 256 + threadIdx.x;            // 65536
  int b = m >> 10, dt = m & 1023;
  const float* p = vid + (size_t)b * (C_ * DT_) + dt;
  float s = 0.f;
  for (int c = 0; c < C_; ++c) s += p[(size_t)c * DT_];
  vmean[m] = s * (1.0f / C_);
}

__global__ void k_wcol(const float* __restrict__ Wv, float* __restrict__ wcol) {
  int n = blockIdx.x * 256 + threadIdx.x;            // 1024
  int k = n >> 8, f = n & 255;
  const float* p = Wv + (size_t)k * (C_ * 256) + f;
  float s = 0.f;
  for (int c = 0; c < C_; ++c) s += p[(size_t)c * 256];
  wcol[n] = s;
}

// ---------------- projection GEMM: v_cat[m][n] = (v-mean).W + b ----------------
__global__ void k_proj_gemm(const float* __restrict__ vid, const float* __restrict__ Wv,
                            const float* __restrict__ bv, const float* __restrict__ vmean,
                            const float* __restrict__ wcol, float* __restrict__ vcat) {
  __shared__ float As[2][32 * LDA_P];                // [k][m]
  __shared__ float Bs[2][32 * LDA_P];                // [k][n]
  const int tid = threadIdx.x, lane = tid & 31, wid = tid >> 5;
  const int wm = wid & 3, wn = wid >> 2;             // 4x2 wave grid
  const int col = lane & 15, rowh = (lane >> 4) * 8, khalf = (lane >> 4) * 2;
  const int nblk = blockIdx.x * 64, mblk = blockIdx.y * 64;
  const int b = mblk >> 10, dt0 = mblk & 1023;
  const float* vbase = vid + (size_t)b * (C_ * DT_) + dt0;

  auto load_tile = [&](int k0, int bufi) {
    for (int i = 0; i < 2; ++i) {                    // A: 512 quads, m-contiguous
      int e = tid + i * 256, kc = e >> 4, mq = (e & 15) * 4;
      cp128(vbase + (size_t)(k0 + kc) * DT_ + mq, &As[bufi][kc * LDA_P + mq]);
    }
    for (int i = 0; i < 2; ++i) {                    // B: n-contiguous
      int e = tid + i * 256, kc = e >> 4, nq = (e & 15) * 4;
      int n = nblk + nq;
      cp128(&Wv[(size_t)(n >> 8) * (C_ * 256) + (size_t)(k0 + kc) * 256 + (n & 255)],
            &Bs[bufi][kc * LDA_P + nq]);
    }
  };

  v8f acc0 = {}, acc1 = {};
  int buf = 0;
  load_tile(0, buf);
  wait_async_lds();
  __syncthreads();
  for (int kb = 0; kb < 32; ++kb) {
    if (kb + 1 < 32) load_tile((kb + 1) * 32, buf ^ 1);
#if HAVE_WMMA_F32
    for (int kk = 0; kk < 8; ++kk) {
      int kl = kk * 4 + khalf;
      v2f a, b0, b1;
      a[0] = As[buf][kl * LDA_P + wm * 16 + col];
      a[1] = As[buf][(kl + 1) * LDA_P + wm * 16 + col];
      b0[0] = Bs[buf][kl * LDA_P + wn * 32 + col];
      b0[1] = Bs[buf][(kl + 1) * LDA_P + wn * 32 + col];
      b1[0] = Bs[buf][kl * LDA_P + wn * 32 + 16 + col];
      b1[1] = Bs[buf][(kl + 1) * LDA_P + wn * 32 + 16 + col];
      acc0 = __builtin_amdgcn_wmma_f32_16x16x4_f32(false, a, false, b0, (short)0, acc0, false, false);
      acc1 = __builtin_amdgcn_wmma_f32_16x16x4_f32(false, a, false, b1, (short)0, acc1, false, false);
    }
#else
    for (int kl = 0; kl < 32; ++kl) {
      float bb0 = Bs[buf][kl * LDA_P + wn * 32 + col];
      float bb1 = Bs[buf][kl * LDA_P + wn * 32 + 16 + col];
      for (int i = 0; i < 8; ++i) {
        float av = As[buf][kl * LDA_P + wm * 16 + i + rowh];
        acc0[i] += av * bb0; acc1[i] += av * bb1;
      }
    }
#endif
    wait_async_lds();
    __syncthreads();
    buf ^= 1;
  }
  for (int i = 0; i < 8; ++i) {
    int m = mblk + wm * 16 + i + rowh;
    int n0 = nblk + wn * 32 + col, n1 = n0 + 16;
    float mu = vmean[m];
    vcat[(size_t)m * N_ + n0] = acc0[i] - mu * wcol[n0] + bv[n0];
    vcat[(size_t)m * N_ + n1] = acc1[i] - mu * wcol[n1] + bv[n1];
  }
}

// reciprocal L2 norm per v_cat row (one wave32 per row)
__global__ void k_rvnorm(const float* __restrict__ vcat, float* __restrict__ rvn) {
  int wid = threadIdx.x >> 5, lane = threadIdx.x & 31;
  int row = blockIdx.x * 8 + wid;
  const float* p = vcat + (size_t)row * N_;
  float s = 0.f;
  for (int j = 0; j < 32; ++j) { float v = p[lane + j * 32]; s += v * v; }
  for (int off = 16; off > 0; off >>= 1) s += __shfl_xor(s, off, 32);
  if (lane == 0) rvn[row] = 1.0f / fmaxf(sqrtf(s), 1e-8f);
}

// sentence path: center, project (4x C->F4), normalize -> sn[s][n]
__global__ void k_sent(const float* __restrict__ sent, const float* __restrict__ Ws,
                       const float* __restrict__ bs, float* __restrict__ sn) {
  __shared__ float sh[1024];
  __shared__ float sc[1024];
  __shared__ float red[256];
  int s = blockIdx.x, tid = threadIdx.x;
  for (int j = 0; j < 4; ++j) sh[tid + j * 256] = sent[(size_t)s * C_ + tid + j * 256];
  __syncthreads();
  float loc = 0.f;
  for (int j = 0; j < 4; ++j) loc += sh[tid + j * 256];
  red[tid] = loc; __syncthreads();
  for (int off = 128; off > 0; off >>= 1) { if (tid < off) red[tid] += red[tid + off]; __syncthreads(); }
  float mean = red[0] * (1.0f / C_);
  __syncthreads();
  for (int j = 0; j < 4; ++j) {
    int n = tid + j * 256;                           // k = j, f = tid
    float a = bs[n];
    const float* w = Ws + (size_t)j * (C_ * 256) + tid;
    for (int c = 0; c < C_; ++c) a += (sh[c] - mean) * w[(size_t)c * 256];
    sc[n] = a;
  }
  __syncthreads();
  loc = 0.f;
  for (int j = 0; j < 4; ++j) { float v = sc[tid + j * 256]; loc += v * v; }
  red[tid] = loc; __syncthreads();
  for (int off = 128; off > 0; off >>= 1) { if (tid < off) red[tid] += red[tid + off]; __syncthreads(); }
  float rn = 1.0f / fmaxf(sqrtf(red[0]), 1e-8f);
  for (int j = 0; j < 4; ++j) { int n = tid + j * 256; sn[(size_t)s * N_ + n] = sc[n] * rn; }
}

// ---------------- similarity GEMM: sim[s][m] = rvn[m] * sum_n vcat[m][n]*sn[s][n] ----------------
// Both operands are K-contiguous in memory -> LDS tiles [row][k], b64 fragment loads.
__global__ void k_sim_gemm(const float* __restrict__ sn, const float* __restrict__ vcat,
                           const float* __restrict__ rvn, const int* __restrict__ mask,
                           float* __restrict__ sim) {
  __shared__ float As[2][64 * LDA_S];                // [s][k]
  __shared__ float Bs[2][64 * LDA_S];                // [m][k]
  const int tid = threadIdx.x, lane = tid & 31, wid = tid >> 5;
  const int wm = wid & 3, wn = wid >> 2;
  const int col = lane & 15, rowh = (lane >> 4) * 8, khalf = (lane >> 4) * 2;
  const int nblk = blockIdx.x * 64;                  // along m (65536)

  auto load_tile = [&](int k0, int bufi) {
    for (int i = 0; i < 2; ++i) {                    // A: 512 quads, k-contiguous
      int e = tid + i * 256, sq = e >> 3, kq = (e & 7) * 4;
      cp128(&sn[(size_t)sq * N_ + k0 + kq], &As[bufi][sq * LDA_S + kq]);
    }
    for (int i = 0; i < 2; ++i) {                    // B: k-contiguous
      int e = tid + i * 256, nn = e >> 3, kq = (e & 7) * 4;
      cp128(&vcat[(size_t)(nblk + nn) * N_ + k0 + kq], &Bs[bufi][nn * LDA_S + kq]);
    }
  };

  v8f acc0 = {}, acc1 = {};
  int buf = 0;
  load_tile(0, buf);
  wait_async_lds();
  __syncthreads();
  for (int kb = 0; kb < 32; ++kb) {
    if (kb + 1 < 32) load_tile((kb + 1) * 32, buf ^ 1);
#if HAVE_WMMA_F32
    for (int kk = 0; kk < 8; ++kk) {
      int kl = kk * 4 + khalf;
      v2f a, b0, b1;
      a[0] = As[buf][(wm * 16 + col) * LDA_S + kl];
      a[1] = As[buf][(wm * 16 + col) * LDA_S + kl + 1];
      b0[0] = Bs[buf][(wn * 32 + col) * LDA_S + kl];
      b0[1] = Bs[buf][(wn * 32 + col) * LDA_S + kl + 1];
      b1[0] = Bs[buf][(wn * 32 + 16 + col) * LDA_S + kl];
      b1[1] = Bs[buf][(wn * 32 + 16 + col) * LDA_S + kl + 1];
      acc0 = __builtin_amdgcn_wmma_f32_16x16x4_f32(false, a, false, b0, (short)0, acc0, false, false);
      acc1 = __builtin_amdgcn_wmma_f32_16x16x4_f32(false, a, false, b1, (short)0, acc1, false, false);
    }
#else
    for (int kl = 0; kl < 32; ++kl) {
      for (int i = 0; i < 8; ++i) {
        float av = As[buf][(wm * 16 + i + rowh) * LDA_S + kl];
        acc0[i] += av * Bs[buf][(wn * 32 + col) * LDA_S + kl];
        acc1[i] += av * Bs[buf][(wn * 32 + 16 + col) * LDA_S + kl];
      }
    }
#endif
    wait_async_lds();
    __syncthreads();
    buf ^= 1;
  }
  for (int i = 0; i < 8; ++i) {
    int srow = wm * 16 + i + rowh;
    int m0 = nblk + wn * 32 + col, m1 = m0 + 16;
    float v0 = acc0[i] * rvn[m0];
    float v1 = acc1[i] * rvn[m1];
    if (mask[m0] == 0) v0 = -INFINITY;
    if (mask[m1] == 0) v1 = -INFINITY;
    sim[(size_t)srow * M_ + m0] = v0;
    sim[(size_t)srow * M_ + m1] = v1;
  }
}

// positive_map[b,dt] = sim[s=b][b*1024+dt]
__global__ void k_posmap(const float* __restrict__ sim, float* __restrict__ out) {
  int i = blockIdx.x * 256 + threadIdx.x;            // 65536
  int b = i >> 10;
  out[1 + i] = sim[(size_t)b * M_ + i];
}

// top-50 (descending, ties -> lower index) per (s,b) row of 1024
__global__ void k_topk(const float* __restrict__ sim, float* __restrict__ vals, int* __restrict__ idx) {
  __shared__ float sv[1024];
  __shared__ float rv[256];
  __shared__ int ri[256];
  int r = blockIdx.x, tid = threadIdx.x;
  int s = r >> 6, b = r & 63;
  const float* row = sim + (size_t)s * M_ + (size_t)b * 1024;
  for (int j = 0; j < 4; ++j) sv[tid + j * 256] = row[tid + j * 256];
  __syncthreads();
  for (int it = 0; it < VN; ++it) {
    float bv = -INFINITY; int bi = 1 << 30;
    for (int j = 0; j < 4; ++j) {
      int e = tid + j * 256; float v = sv[e];
      if (v > bv || (v == bv && e < bi)) { bv = v; bi = e; }
    }
    rv[tid] = bv; ri[tid] = bi;
    __syncthreads();
    for (int off = 128; off > 0; off >>= 1) {
      if (tid < off) {
        float ov = rv[tid + off]; int oi = ri[tid + off];
        if (ov > rv[tid] || (ov == rv[tid] && oi < ri[tid])) { rv[tid] = ov; ri[tid] = oi; }
      }
      __syncthreads();
    }
    if (tid == 0) {
      vals[(size_t)r * VN + it] = rv[0];
      idx[(size_t)r * VN + it] = ri[0];
      sv[ri[0]] = -INFINITY;
    }
    __syncthreads();
  }
}

// penalty (strict lower tri: l>k), w = exp(-p/lam), score = mean(vals*w)
__global__ void k_scores(const float* __restrict__ vals, const int* __restrict__ idx,
                         const float* __restrict__ iou, const float* __restrict__ lam,
                         float* __restrict__ scores) {
  __shared__ int sidx[VN];
  __shared__ float red[64];
  int r = blockIdx.x, tid = threadIdx.x;             // 64 threads
  if (tid < VN) sidx[tid] = idx[(size_t)r * VN + tid];
  __syncthreads();
  float contrib = 0.f;
  if (tid < VN) {
    int ik = sidx[tid];
    float p = 0.f;
    for (int l = tid + 1; l < VN; ++l) {
      float u = iou[(size_t)sidx[l] * 1024 + ik];
      p += u * u;
    }
    contrib = vals[(size_t)r * VN + tid] * __expf(-p / lam[0]);
  }
  red[tid] = contrib; __syncthreads();
  for (int off = 32; off > 0; off >>= 1) { if (tid < off) red[tid] += red[tid + off]; __syncthreads(); }
  if (tid == 0) scores[r] = red[0] * (1.0f / VN);
}

// max-violation contrastive loss
__global__ void k_loss(const float* __restrict__ scores, float* __restrict__ out) {
  __shared__ float diag[64];
  __shared__ float r1[64];
  __shared__ float r2[64];
  int t = threadIdx.x;                               // 64 threads
  diag[t] = scores[t * 64 + t];
  __syncthreads();
  float cs = 0.f, cim = 0.f;
  for (int j = 0; j < 64; ++j) {
    if (j != t) {
      float v = 0.2f + scores[t * 64 + j] - diag[t];
      cs = fmaxf(cs, fmaxf(v, 0.f));
      float u = 0.2f + scores[j * 64 + t] - diag[t];
      cim = fmaxf(cim, fmaxf(u, 0.f));
    }
  }
  r1[t] = cs; r2[t] = cim;
  __syncthreads();
  for (int off = 32; off > 0; off >>= 1) { if (t < off) { r1[t] += r1[t + off]; r2[t] += r2[t + off]; } __syncthreads(); }
  if (t == 0) out[0] = (r1[0] + r2[0]) * (1.0f / 64.0f);
}

extern "C" void kernel_launch(void* const* d_in, const int* in_sizes, int n_in,
                              void* d_out, int out_size, void* d_ws, size_t ws_size,
                              hipStream_t stream) {
  const float* vid  = (const float*)d_in[0];
  const float* sent = (const float*)d_in[1];
  const float* lam  = (const float*)d_in[2];
  const int*   mask = (const int*)d_in[3];
  const float* iou  = (const float*)d_in[4];
  const float* Wv   = (const float*)d_in[6];
  const float* bv   = (const float*)d_in[7];
  const float* Ws   = (const float*)d_in[8];
  const float* bs   = (const float*)d_in[9];
  float* out = (float*)d_out;

  float* w = (float*)d_ws;
  float* vmean  = w;                       // 65536
  float* wcol   = vmean + 65536;           // 1024
  float* rvn    = wcol + 1024;             // 65536
  float* sn     = rvn + 65536;             // 65536
  float* scores = sn + 65536;              // 4096
  float* vals   = scores + 4096;           // 204800
  int*   idx    = (int*)(vals + 204800);   // 204800
  float* vcat   = (float*)(idx + 204800);  // 67108864
  float* sim    = vcat + (size_t)M_ * N_;  // 4194304

  k_vmean<<<256, 256, 0, stream>>>(vid, vmean);
  k_wcol<<<4, 256, 0, stream>>>(Wv, wcol);
  k_proj_gemm<<<dim3(16, 1024), 256, 0, stream>>>(vid, Wv, bv, vmean, wcol, vcat);
  k_rvnorm<<<8192, 256, 0, stream>>>(vcat, rvn);
  k_sent<<<64, 256, 0, stream>>>(sent, Ws, bs, sn);
  k_sim_gemm<<<1024, 256, 0, stream>>>(sn, vcat, rvn, mask, sim);
  k_posmap<<<256, 256, 0, stream>>>(sim, out);
  k_topk<<<4096, 256, 0, stream>>>(sim, vals, idx);
  k_scores<<<4096, 64, 0, stream>>>(vals, idx, iou, lam, scores);
  k_loss<<<1, 64, 0, stream>>>(scores, out);
}